// TransactionLayer_54056458387499
// MI455X (gfx1250) — compile-verified
//
#include <hip/hip_runtime.h>
#include <hip/hip_bf16.h>
#include <cstdint>
#include <math.h>

// Problem shape (from reference): in [32, 224, 224, 128] f32, out [32, 128] f32.
#define BATCH   32
#define SPATIAL (224 * 224)
#define CHANS   128

// ---------------------------------------------------------------------------
// Seed output with -inf each call (harness poisons d_out; atomic-max needs a
// proper identity element).
// ---------------------------------------------------------------------------
__global__ void TransactionLayer_init_kernel(float* __restrict__ out, int n) {
    int i = blockIdx.x * blockDim.x + threadIdx.x;
    if (i < n) out[i] = -INFINITY;
}

// ---------------------------------------------------------------------------
// Helpers
// ---------------------------------------------------------------------------
__device__ __forceinline__ float4 max4(float4 a, float4 b) {
    a.x = fmaxf(a.x, b.x);
    a.y = fmaxf(a.y, b.y);
    a.z = fmaxf(a.z, b.z);
    a.w = fmaxf(a.w, b.w);
    return a;
}

// gfx1250 native float-max atomic (GLOBAL_ATOMIC_MAX_NUM_F32, no-return form),
// device scope so the RMW happens coherently in L2 across all WGPs.
__device__ __forceinline__ void atomic_max_num_f32(float* addr, float val) {
    asm volatile("global_atomic_max_num_f32 %0, %1, off scope:SCOPE_DEV"
                 :
                 : "v"((unsigned long long)(uintptr_t)addr), "v"(val)
                 : "memory");
}

// ---------------------------------------------------------------------------
// Streaming max-reduction over spatial axes.
//   grid  = (SPATIAL_CHUNKS, BATCH), block = 256 (8 wave32s)
//   lane mapping: c4 = tid & 31  -> float4 channel group (4 consecutive chans)
//                 sg = tid >> 5  -> spatial sub-slot (8 positions / iteration)
// Each iteration a block reads 8 positions * 128 chans * 4B = 4 KB contiguous.
// ---------------------------------------------------------------------------
__global__ __launch_bounds__(256) void TransactionLayer_54056458387499_kernel(
        const float* __restrict__ in, float* __restrict__ out) {
    const int tid = threadIdx.x;
    const int c4  = tid & 31;   // channel group index (float4 granularity)
    const int sg  = tid >> 5;   // 0..7
    const int b   = blockIdx.y;

    // float4-typed base for this (batch, channel-group); element index for
    // spatial position s is s*32 (32 float4s = 128 channels per position).
    const float4* base =
        (const float4*)(in + (size_t)b * (size_t)SPATIAL * CHANS) + c4;

    const long long stride = (long long)gridDim.x * 8;
    long long s = (long long)blockIdx.x * 8 + sg;

    float4 acc0 = make_float4(-INFINITY, -INFINITY, -INFINITY, -INFINITY);
    float4 acc1 = acc0;

    // Two independent accumulators -> two global_load_b128 in flight per iter.
    for (; s + stride < SPATIAL; s += 2 * stride) {
        float4 v0 = base[s * 32];
        float4 v1 = base[(s + stride) * 32];
        acc0 = max4(acc0, v0);
        acc1 = max4(acc1, v1);
    }
    if (s < SPATIAL) {
        acc0 = max4(acc0, base[s * 32]);
    }
    acc0 = max4(acc0, acc1);

    // Block-level reduction across the 8 spatial sub-slots via LDS.
    __shared__ float4 red[256];
    red[tid] = acc0;
    __syncthreads();

    if (sg == 0) {
        float4 m = red[c4];
#pragma unroll
        for (int k = 1; k < 8; ++k) {
            m = max4(m, red[c4 + 32 * k]);
        }
        float* o = out + (size_t)b * CHANS + c4 * 4;
        atomic_max_num_f32(o + 0, m.x);
        atomic_max_num_f32(o + 1, m.y);
        atomic_max_num_f32(o + 2, m.z);
        atomic_max_num_f32(o + 3, m.w);
    }
}

// ---------------------------------------------------------------------------
// Launch
// ---------------------------------------------------------------------------
extern "C" void kernel_launch(void* const* d_in, const int* in_sizes, int n_in,
                              void* d_out, int out_size, void* d_ws, size_t ws_size,
                              hipStream_t stream) {
    (void)in_sizes; (void)n_in; (void)d_ws; (void)ws_size;

    const float* in = (const float*)d_in[0];
    float* out = (float*)d_out;

    // Seed output with -inf (deterministic every call).
    {
        int threads = 256;
        int blocks  = (out_size + threads - 1) / threads;
        TransactionLayer_init_kernel<<<blocks, threads, 0, stream>>>(out, out_size);
    }

    // 256 spatial chunks x 32 batches = 8192 blocks of 8 wave32s.
    dim3 grid(256, BATCH, 1);
    dim3 block(256, 1, 1);
    TransactionLayer_54056458387499_kernel<<<grid, block, 0, stream>>>(in, out);
}